// FreTransferInv_33071248179687
// MI455X (gfx1250) — compile-verified
//
#include <hip/hip_runtime.h>

// Inverse 2x2 Haar-interleave (FreTransferInv):
//   in  x : (8, 16, 512, 512) f32   -> a=ch, b=ch+4, c=ch+8, d=ch+12 subbands
//   out   : (8,  4, 1024,1024) f32
// Pure streaming op (128 MiB in + 128 MiB out, ~0.37 flop/B) -> HBM-bound,
// floor ~11.5us @ 23.3 TB/s. Strategy: async global->LDS b128 stream-in
// (ASYNCcnt), butterfly in VALU, non-temporal b128 stream-out.

typedef float v4f __attribute__((ext_vector_type(4)));

#define BLOCK 256

__device__ __forceinline__ unsigned lds_off(const void* p) {
    // generic pointer to LDS: low 32 bits are the LDS byte offset
    return (unsigned)(size_t)p;
}

__global__ __launch_bounds__(BLOCK) void
fretransferinv_kernel(const float* __restrict__ x,
                      float* __restrict__ out,
                      const float* __restrict__ g0p,
                      const float* __restrict__ g1p)
{
    // 4 regions (a,b,c,d), one 16B slot per thread per region.
    // 16B lane stride -> conflict-free ds_load_b128.
    __shared__ v4f smem4[4 * BLOCK];          // 16 KB

    const unsigned t   = threadIdx.x;
    const unsigned gid = blockIdx.x * BLOCK + t;

    // gid -> (n, ch, i, jb): jb in [0,128), i in [0,512), ch in [0,4), n in [0,8)
    const unsigned jb = gid & 127u;
    unsigned tmp      = gid >> 7;
    const unsigned i  = tmp & 511u;
    tmp >>= 9;
    const unsigned ch = tmp & 3u;
    const unsigned n  = tmp >> 2;

    const size_t plane   = 512u * 512u;                       // input subband plane
    const size_t base_in = ((size_t)(n * 16u + ch)) * plane
                         + (size_t)i * 512u + (size_t)jb * 4u;

    const float* pa = x + base_in;                // lls
    const float* pb = pa + 4u  * plane;           // hls
    const float* pc = pa + 8u  * plane;           // lhs
    const float* pd = pa + 12u * plane;           // hhs

    const unsigned la = lds_off(&smem4[0 * BLOCK + t]);
    const unsigned lb = lds_off(&smem4[1 * BLOCK + t]);
    const unsigned lc = lds_off(&smem4[2 * BLOCK + t]);
    const unsigned ld = lds_off(&smem4[3 * BLOCK + t]);

    // CDNA5 async data mover: global -> LDS, tracked by ASYNCcnt (no VGPR
    // return registers held in flight).
    asm volatile("global_load_async_to_lds_b128 %0, %1, off"
                 :: "v"(la), "v"(pa) : "memory");
    asm volatile("global_load_async_to_lds_b128 %0, %1, off"
                 :: "v"(lb), "v"(pb) : "memory");
    asm volatile("global_load_async_to_lds_b128 %0, %1, off"
                 :: "v"(lc), "v"(pc) : "memory");
    asm volatile("global_load_async_to_lds_b128 %0, %1, off"
                 :: "v"(ld), "v"(pd) : "memory");

    // Uniform filter taps (scalar loads, broadcast)
    const float g00 = g0p[0], g01 = g0p[1];
    const float g10 = g1p[0], g11 = g1p[1];

    asm volatile("s_wait_asynccnt 0x0" ::: "memory");

    // Each thread reads back only its own slots -> no workgroup barrier needed.
    const v4f a = smem4[0 * BLOCK + t];
    const v4f b = smem4[1 * BLOCK + t];
    const v4f c = smem4[2 * BLOCK + t];
    const v4f d = smem4[3 * BLOCK + t];

    // Row-direction butterfly (axis=2 interleave)
    const v4f lo0 = g00 * a + g10 * b;   // output row 2i
    const v4f lo1 = g01 * a + g11 * b;   // output row 2i+1
    const v4f hi0 = g00 * c + g10 * d;
    const v4f hi1 = g01 * c + g11 * d;

    // Column-direction butterfly (axis=3 interleave)
    const v4f r0e = g00 * lo0 + g10 * hi0;   // row 2i,   even cols
    const v4f r0o = g01 * lo0 + g11 * hi0;   // row 2i,   odd  cols
    const v4f r1e = g00 * lo1 + g10 * hi1;   // row 2i+1, even cols
    const v4f r1o = g01 * lo1 + g11 * hi1;   // row 2i+1, odd  cols

    // Interleave even/odd columns into contiguous 8-float runs
    const v4f o00 = {r0e.x, r0o.x, r0e.y, r0o.y};
    const v4f o01 = {r0e.z, r0o.z, r0e.w, r0o.w};
    const v4f o10 = {r1e.x, r1o.x, r1e.y, r1o.y};
    const v4f o11 = {r1e.z, r1o.z, r1e.w, r1o.w};

    const size_t oplane   = 1024u * 1024u;
    const size_t base_out = ((size_t)(n * 4u + ch)) * oplane
                          + (size_t)(2u * i) * 1024u + (size_t)jb * 8u;
    float* q0 = out + base_out;          // row 2i
    float* q1 = q0 + 1024u;              // row 2i+1

    // Write-once 128 MiB stream: non-temporal stores keep it out of L2.
    __builtin_nontemporal_store(o00, (v4f*)(q0));
    __builtin_nontemporal_store(o01, (v4f*)(q0 + 4));
    __builtin_nontemporal_store(o10, (v4f*)(q1));
    __builtin_nontemporal_store(o11, (v4f*)(q1 + 4));
}

extern "C" void kernel_launch(void* const* d_in, const int* in_sizes, int n_in,
                              void* d_out, int out_size, void* d_ws, size_t ws_size,
                              hipStream_t stream) {
    const float* x  = (const float*)d_in[0];
    const float* g0 = (const float*)d_in[1];
    const float* g1 = (const float*)d_in[2];
    float* out = (float*)d_out;

    // Each thread consumes 16 input elements (4 from each subband).
    const long total_threads = (long)in_sizes[0] / 16;   // 8*4*512*128 = 2,097,152
    const int  blocks        = (int)(total_threads / BLOCK);  // 8192

    fretransferinv_kernel<<<blocks, BLOCK, 0, stream>>>(x, out, g0, g1);
}